// sampleGCN_15556371546325
// MI455X (gfx1250) — compile-verified
//
#include <hip/hip_runtime.h>
#include <cstdint>

typedef __attribute__((ext_vector_type(2))) float v2f;
typedef __attribute__((ext_vector_type(8))) float v8f;

#define SSCOPE 128
#define FDIM   602
#define HDIM   256
#define ODIM   41
#define KSAMP  10

// ---------------- JAX threefry2x32 (20 rounds) ----------------
__device__ __forceinline__ uint32_t rotl32(uint32_t x, int r) {
  return (x << r) | (x >> (32 - r));
}

__device__ __forceinline__ void threefry2x32(uint32_t k0, uint32_t k1,
                                             uint32_t x0, uint32_t x1,
                                             uint32_t* o0, uint32_t* o1) {
  uint32_t ks2 = 0x1BD11BDAu ^ k0 ^ k1;
  x0 += k0; x1 += k1;
#define TF_R(r) { x0 += x1; x1 = rotl32(x1, (r)); x1 ^= x0; }
  TF_R(13) TF_R(15) TF_R(26) TF_R(6)
  x0 += k1;  x1 += ks2 + 1u;
  TF_R(17) TF_R(29) TF_R(16) TF_R(24)
  x0 += ks2; x1 += k0 + 2u;
  TF_R(13) TF_R(15) TF_R(26) TF_R(6)
  x0 += k0;  x1 += k1 + 3u;
  TF_R(17) TF_R(29) TF_R(16) TF_R(24)
  x0 += k1;  x1 += ks2 + 4u;
  TF_R(13) TF_R(15) TF_R(26) TF_R(6)
  x0 += ks2; x1 += k0 + 5u;
#undef TF_R
  *o0 = x0; *o1 = x1;
}

// hdr layout (16 dwords): [0..2] = softmax(a), [4..5] = sk(step0), [6..7] = sk(step1)
__global__ void gcn_prep_kernel(const float* __restrict__ sample_a, float* __restrict__ hdr) {
  if (threadIdx.x != 0) return;
  float a0 = sample_a[0], a1 = sample_a[1], a2 = sample_a[2];
  float m = fmaxf(a0, fmaxf(a1, a2));
  float e0 = expf(a0 - m), e1 = expf(a1 - m), e2 = expf(a2 - m);
  float s = e0 + e1 + e2;
  hdr[0] = e0 / s; hdr[1] = e1 / s; hdr[2] = e2 / s;
  uint32_t* h = (uint32_t*)hdr;
  // key = (0, 42);  split: counts iota(4) -> pairs (0,2) and (1,3)
  uint32_t k0 = 0u, k1 = 42u;
  uint32_t y00, y10, y01, y11;
  threefry2x32(k0, k1, 0u, 2u, &y00, &y10);
  threefry2x32(k0, k1, 1u, 3u, &y01, &y11);
  // key' = (y00, y01), sk0 = (y10, y11)
  h[4] = y10; h[5] = y11;
  uint32_t nk0 = y00, nk1 = y01;
  threefry2x32(nk0, nk1, 0u, 2u, &y00, &y10);
  threefry2x32(nk0, nk1, 1u, 3u, &y01, &y11);
  h[6] = y10; h[7] = y11;   // sk1
}

// --------- async global->LDS helpers (CDNA5 ASYNCcnt path, inline asm) ---------
// LDS operand = low 32 bits of the generic pointer (flat->LDS aperture keeps the
// byte offset in addr[31:0], ISA 10.2). 16 bytes per lane per issue.
__device__ __forceinline__ void async_load_lds_b128(void* lds_dst, const float* gsrc) {
  uint32_t ldsa = (uint32_t)(uintptr_t)lds_dst;
  uint64_t ga = (uint64_t)(uintptr_t)gsrc;
  asm volatile("global_load_async_to_lds_b128 %0, %1, off"
               :: "v"(ldsa), "v"(ga) : "memory");
}
__device__ __forceinline__ void wait_asynccnt0() {
  asm volatile("s_wait_asynccnt 0" ::: "memory");
}

// ---------------- fp32 WMMA GEMM ----------------
// C[M,N] = op( alpha * gather(A,rowIdx) @ B(^T) )  (+C if accum, relu optional)
// block = 128 threads (4 waves); tile M=16, N=64 (wave w -> N sub-tile w*16); K chunk 32.
#define TKC 32
__global__ void gcn_gemm_wmma_f32(const float* __restrict__ A, const int* __restrict__ rowIdx, int lda,
                                  const float* __restrict__ B, int ldb, int transB,
                                  float* __restrict__ C, int ldc,
                                  int M, int N, int K,
                                  const float* __restrict__ alpha_ptr, int accum, int do_relu) {
  __shared__ float As[16][33];
  __shared__ float Bs[TKC][65];
  int m0 = blockIdx.y * 16;
  int n0 = blockIdx.x * 64;
  int tid = threadIdx.x;
  int wave = tid >> 5;
  int lane = tid & 31;
  int hi = lane >> 4;        // 0/1
  int ml = lane & 15;        // m for A frags / n-within-subtile for B frags
  v8f acc = {};

  // per-thread A staging coordinates (fixed): row rA, 4-float group kgA
  int rA = tid >> 3;          // 0..15
  int kgA = (tid & 7);        // group of 4 k
  const float* arowA;
  {
    int gm = m0 + rA;         // M is a multiple of 16 for all call sites
    int ridx = rowIdx ? rowIdx[gm] : gm;
    arowA = A + (size_t)ridx * (size_t)lda;
  }

  for (int k0 = 0; k0 < K; k0 += TKC) {
    bool fullK = (k0 + TKC <= K);
    bool asyncB = fullK && (!transB) && (n0 + 64 <= N);
    // --- stage A tile: 16 rows x 32 k ---
    if (fullK) {
      async_load_lds_b128(&As[rA][kgA * 4], arowA + k0 + kgA * 4);
    } else {
#pragma unroll
      for (int u = 0; u < 4; ++u) {
        int gk = k0 + kgA * 4 + u;
        As[rA][kgA * 4 + u] = (gk < K) ? arowA[gk] : 0.0f;
      }
    }
    // --- stage B tile: 32 k x 64 n ---
    if (asyncB) {
      int ng = tid & 15;                 // 4-float group in n
#pragma unroll
      for (int j = 0; j < 4; ++j) {
        int kk = (tid >> 4) + j * 8;     // 0..31
        async_load_lds_b128(&Bs[kk][ng * 4],
                            B + (size_t)(k0 + kk) * (size_t)ldb + n0 + ng * 4);
      }
    } else if (!transB) {
      int nn = (tid & 15) * 4;
#pragma unroll
      for (int j = 0; j < 4; ++j) {
        int kk = (tid >> 4) + j * 8;     // 0..31
        int gk = k0 + kk;
#pragma unroll
        for (int u = 0; u < 4; ++u) {
          int gn = n0 + nn + u;
          Bs[kk][nn + u] = (gk < K && gn < N) ? B[(size_t)gk * (size_t)ldb + gn] : 0.0f;
        }
      }
    } else {
      // B_eff[k][n] = B[n*ldb + k]  (read W rows contiguously, store transposed)
      int nn = tid >> 1;             // 0..63
      int kk = (tid & 1) * 16;       // 0 / 16
      int gn = n0 + nn;
#pragma unroll
      for (int u = 0; u < 16; ++u) {
        int gk = k0 + kk + u;
        Bs[kk + u][nn] = (gk < K && gn < N) ? B[(size_t)gn * (size_t)ldb + gk] : 0.0f;
      }
    }
    if (fullK) wait_asynccnt0();
    __syncthreads();
    // --- 8 x V_WMMA_F32_16X16X4_F32 per chunk ---
#pragma unroll
    for (int kk = 0; kk < TKC; kk += 4) {
      v2f av, bv;
      av.x = As[ml][kk + 2 * hi];
      av.y = As[ml][kk + 2 * hi + 1];
      bv.x = Bs[kk + 2 * hi][wave * 16 + ml];
      bv.y = Bs[kk + 2 * hi + 1][wave * 16 + ml];
      acc = __builtin_amdgcn_wmma_f32_16x16x4_f32(
          /*neg_a=*/false, av, /*neg_b=*/false, bv,
          /*c_mod=*/(short)0, acc, /*reuse_a=*/false, /*reuse_b=*/false);
    }
    __syncthreads();
  }

  float alpha = alpha_ptr ? *alpha_ptr : 1.0f;
  int n = n0 + wave * 16 + ml;
  if (n < N) {
#pragma unroll
    for (int i = 0; i < 8; ++i) {
      int m = m0 + i + 8 * hi;
      size_t off = (size_t)m * (size_t)ldc + n;
      float v = acc[i] * alpha;
      if (accum) v += C[off];
      if (do_relu) v = fmaxf(v, 0.0f);
      C[off] = v;
    }
  }
}

// ---------------- attention scores: att[n,s] = relu(q[n]·F[nbr] + a2*w) + 1e-9 ----------------
__global__ void gcn_att_kernel(const float* __restrict__ q, const int* __restrict__ frontier,
                               const float* __restrict__ feature, const int* __restrict__ edge,
                               const float* __restrict__ weight, const float* __restrict__ hdr,
                               float* __restrict__ att) {
  int n = blockIdx.x;
  int tid = threadIdx.x;              // 256
  int wave = tid >> 5, lane = tid & 31;
  __shared__ float qs[FDIM];
  int node = frontier[n];
  for (int i = tid; i < FDIM; i += 256) qs[i] = q[(size_t)n * FDIM + i];
  __syncthreads();
  float a2 = hdr[2];
  const int*   erow = edge   + (size_t)node * SSCOPE;
  const float* wrow = weight + (size_t)node * SSCOPE;
  for (int s = wave; s < SSCOPE; s += 8) {
    const float* frow = feature + (size_t)erow[s] * FDIM;
    // prefetch the next gathered row for this wave while we reduce this one
    if (s + 8 < SSCOPE) {
      const float* nxt = feature + (size_t)erow[s + 8] * FDIM;
      __builtin_prefetch(nxt + lane * 20, 0, 0);   // 80B stride x 32 lanes covers 2408B row
    }
    float acc = 0.0f;
    for (int i = lane; i < FDIM; i += 32) acc += qs[i] * frow[i];
#pragma unroll
    for (int off = 16; off > 0; off >>= 1) acc += __shfl_down(acc, off, 32);
    if (lane == 0)
      att[(size_t)n * SSCOPE + s] = fmaxf(acc + a2 * wrow[s], 0.0f) + 1e-9f;
  }
}

// ---------------- Gumbel-argmax categorical sampling (JAX threefry bits) ----------------
// block = 320 threads = 10 waves; wave k draws sample k for node blockIdx.x.
__global__ void gcn_sample_kernel(const float* __restrict__ att, const int* __restrict__ frontier,
                                  const int* __restrict__ edge, const float* __restrict__ weight,
                                  const uint32_t* __restrict__ hdr_u, int key_sel, int n_nodes,
                                  float* __restrict__ adj, int* __restrict__ next_frontier) {
  int n = blockIdx.x;
  int tid = threadIdx.x;
  int wave = tid >> 5, lane = tid & 31;   // wave = sample index k (0..9)
  uint32_t k0 = hdr_u[4 + 2 * key_sel];
  uint32_t k1 = hdr_u[5 + 2 * key_sel];
  __shared__ float logit_s[SSCOPE];
  __shared__ float wsel[KSAMP];
  __shared__ int   nsel[KSAMP];
  for (int i = tid; i < SSCOPE; i += 320)
    logit_s[i] = logf(att[(size_t)n * SSCOPE + i]);
  __syncthreads();

  uint32_t size = (uint32_t)KSAMP * (uint32_t)n_nodes * (uint32_t)SSCOPE;
  uint32_t half = size >> 1;
  const float tiny = 1.1754944e-38f;
  float best = -INFINITY; int bestIdx = 0;
  for (int s = lane; s < SSCOPE; s += 32) {
    uint32_t idx = ((uint32_t)wave * (uint32_t)n_nodes + (uint32_t)n) * (uint32_t)SSCOPE + (uint32_t)s;
    uint32_t x0, x1; int use_hi;
    if (idx < half) { x0 = idx; x1 = idx + half; use_hi = 0; }
    else            { x0 = idx - half; x1 = idx; use_hi = 1; }
    uint32_t y0, y1;
    threefry2x32(k0, k1, x0, x1, &y0, &y1);
    uint32_t bits = use_hi ? y1 : y0;
    float u = __uint_as_float((bits >> 9) | 0x3f800000u) - 1.0f;
    u = fmaxf(tiny, u * (1.0f - tiny) + tiny);
    float g = -logf(-logf(u));
    float v = logit_s[s] + g;
    if (v > best) { best = v; bestIdx = s; }
  }
#pragma unroll
  for (int off = 16; off > 0; off >>= 1) {
    float ob = __shfl_xor(best, off, 32);
    int   oi = __shfl_xor(bestIdx, off, 32);
    if (ob > best || (ob == best && oi < bestIdx)) { best = ob; bestIdx = oi; }
  }
  if (lane == 0) {
    int node = frontier[n];
    wsel[wave] = weight[(size_t)node * SSCOPE + bestIdx];
    nsel[wave] = edge[(size_t)node * SSCOPE + bestIdx];
  }
  __syncthreads();
  if (tid < KSAMP) {
    float sum = 0.0f;
#pragma unroll
    for (int k = 0; k < KSAMP; ++k) sum += wsel[k];
    adj[(size_t)n * KSAMP + tid] = wsel[tid] / (sum + 1e-9f);
    next_frontier[n * KSAMP + tid] = nsel[tid];
  }
}

// ---------------- weighted neighbor aggregation: H[n] = sum_k adj[n,k] * X[row(n,k)] ----------------
__global__ void gcn_agg_kernel(const float* __restrict__ adj, const int* __restrict__ rows_idx,
                               const float* __restrict__ X, int dim, float* __restrict__ H) {
  int n = blockIdx.x;
  int tid = threadIdx.x;   // 256
  __shared__ float w[KSAMP];
  __shared__ int   rows[KSAMP];
  if (tid < KSAMP) {
    w[tid]    = adj[(size_t)n * KSAMP + tid];
    rows[tid] = rows_idx ? rows_idx[n * KSAMP + tid] : (n * KSAMP + tid);
  }
  __syncthreads();
  for (int f = tid; f < dim; f += 256) {
    float acc = 0.0f;
#pragma unroll
    for (int k = 0; k < KSAMP; ++k)
      acc += w[k] * X[(size_t)rows[k] * (size_t)dim + f];
    H[(size_t)n * (size_t)dim + f] = acc;
  }
}

extern "C" void kernel_launch(void* const* d_in, const int* in_sizes, int n_in,
                              void* d_out, int out_size, void* d_ws, size_t ws_size,
                              hipStream_t stream) {
  const int*   ids      = (const int*)d_in[0];
  const float* feature  = (const float*)d_in[1];
  const int*   edge     = (const int*)d_in[2];
  const float* weight   = (const float*)d_in[3];
  const float* sample_W  = (const float*)d_in[4];
  const float* sample_W2 = (const float*)d_in[5];
  const float* sample_a  = (const float*)d_in[6];
  const float* W1 = (const float*)d_in[7];
  const float* W2 = (const float*)d_in[8];
  float* out = (float*)d_out;

  const int n0 = 256, n1 = 2560;

  float* ws = (float*)d_ws;
  float* hdr = ws;
  uint32_t* hdr_u = (uint32_t*)d_ws;
  size_t off = 16;
  float* S1  = ws + off; off += (size_t)n1 * HDIM;
  float* S2  = ws + off; off += (size_t)n1 * HDIM;
  float* Q   = ws + off; off += (size_t)n1 * FDIM;
  float* ATT = ws + off; off += (size_t)n1 * SSCOPE;
  float* ADJ1 = ws + off; off += (size_t)n0 * KSAMP;
  float* ADJ0 = ws + off; off += (size_t)n1 * KSAMP;
  int* FR1 = (int*)(ws + off); off += n1;
  int* FR2 = (int*)(ws + off); off += n1 * KSAMP;
  float* H0 = Q;   // reuse: q not needed after sampling
  float* X1 = S1;
  float* H1 = S2;
  (void)ws_size; (void)in_sizes; (void)n_in; (void)out_size;

  gcn_prep_kernel<<<1, 32, 0, stream>>>(sample_a, hdr);

  // ---- sampling step 0 (frontier = ids, n=256, layer index 1) ----
  gcn_gemm_wmma_f32<<<dim3(4, 16), 128, 0, stream>>>(feature, ids, FDIM, sample_W,  HDIM, 0,
                                                     S1, HDIM, n0, HDIM, FDIM, nullptr, 0, 0);
  gcn_gemm_wmma_f32<<<dim3(4, 16), 128, 0, stream>>>(feature, ids, FDIM, sample_W2, HDIM, 0,
                                                     S2, HDIM, n0, HDIM, FDIM, nullptr, 0, 0);
  gcn_gemm_wmma_f32<<<dim3(10, 16), 128, 0, stream>>>(S1, nullptr, HDIM, sample_W,  HDIM, 1,
                                                      Q, FDIM, n0, FDIM, HDIM, hdr + 0, 0, 0);
  gcn_gemm_wmma_f32<<<dim3(10, 16), 128, 0, stream>>>(S2, nullptr, HDIM, sample_W2, HDIM, 1,
                                                      Q, FDIM, n0, FDIM, HDIM, hdr + 1, 1, 0);
  gcn_att_kernel<<<n0, 256, 0, stream>>>(Q, ids, feature, edge, weight, hdr, ATT);
  gcn_sample_kernel<<<n0, 320, 0, stream>>>(ATT, ids, edge, weight, hdr_u, 0, n0, ADJ1, FR1);

  // ---- sampling step 1 (frontier = FR1, n=2560, layer index 0) ----
  gcn_gemm_wmma_f32<<<dim3(4, 160), 128, 0, stream>>>(feature, FR1, FDIM, sample_W,  HDIM, 0,
                                                      S1, HDIM, n1, HDIM, FDIM, nullptr, 0, 0);
  gcn_gemm_wmma_f32<<<dim3(4, 160), 128, 0, stream>>>(feature, FR1, FDIM, sample_W2, HDIM, 0,
                                                      S2, HDIM, n1, HDIM, FDIM, nullptr, 0, 0);
  gcn_gemm_wmma_f32<<<dim3(10, 160), 128, 0, stream>>>(S1, nullptr, HDIM, sample_W,  HDIM, 1,
                                                       Q, FDIM, n1, FDIM, HDIM, hdr + 0, 0, 0);
  gcn_gemm_wmma_f32<<<dim3(10, 160), 128, 0, stream>>>(S2, nullptr, HDIM, sample_W2, HDIM, 1,
                                                       Q, FDIM, n1, FDIM, HDIM, hdr + 1, 1, 0);
  gcn_att_kernel<<<n1, 256, 0, stream>>>(Q, FR1, feature, edge, weight, hdr, ATT);
  gcn_sample_kernel<<<n1, 320, 0, stream>>>(ATT, FR1, edge, weight, hdr_u, 1, n1, ADJ0, FR2);

  // ---- aggregation / output ----
  gcn_agg_kernel<<<n1, 256, 0, stream>>>(ADJ0, FR2, feature, FDIM, H0);
  gcn_gemm_wmma_f32<<<dim3(4, 160), 128, 0, stream>>>(H0, nullptr, FDIM, W1, HDIM, 0,
                                                      X1, HDIM, n1, HDIM, FDIM, nullptr, 0, 1);
  gcn_agg_kernel<<<n0, 256, 0, stream>>>(ADJ1, nullptr, X1, HDIM, H1);
  gcn_gemm_wmma_f32<<<dim3(1, 16), 128, 0, stream>>>(H1, nullptr, HDIM, W2, ODIM, 0,
                                                     out, ODIM, n0, ODIM, HDIM, nullptr, 0, 0);
}